// LatentDiffusionExplainer_86474871537940
// MI455X (gfx1250) — compile-verified
//
#include <hip/hip_runtime.h>
#include <hip/hip_bf16.h>

#define D_MODEL 128
#define EDGE_DIM 9
#define HIDDEN 256
#define BATCH 2
#define NNODE 512

typedef __attribute__((ext_vector_type(2))) float v2f;
typedef __attribute__((ext_vector_type(8))) float v8f;
typedef __attribute__((ext_vector_type(4))) unsigned int v4u;
typedef __attribute__((ext_vector_type(4))) int v4i;
typedef __attribute__((ext_vector_type(8))) int v8i;

__device__ __forceinline__ float sigmoidf_(float x) {
    return 1.0f / (1.0f + __expf(-x));
}

// ---------------------------------------------------------------------------
// Tensor Data Mover helpers (CDNA5 D# descriptor, ISA cdna5 ch.8).
// ---------------------------------------------------------------------------
__device__ __forceinline__ unsigned lds_offset(const void* p) {
    // generic -> LDS addrspace(3); numeric value is the byte offset in LDS.
    return (unsigned)(unsigned long long)(__attribute__((address_space(3))) const void*)p;
}

__device__ __forceinline__ v4u tdm_group0(unsigned lds_addr, const void* gptr) {
    const unsigned long long ga = (unsigned long long)gptr;
    v4u g;
    g.x = 1u;                                   // count=1, user-mode descriptor
    g.y = lds_addr;                             // lds_addr (bytes)
    g.z = (unsigned)(ga & 0xFFFFFFFFu);         // global_addr[31:0]
    g.w = (unsigned)((ga >> 32) & 0x01FFFFFFu)  // global_addr[56:32]
        | (2u << 30);                           // type = 2 ("image")
    return g;
}

// 2D tile: data_size=4B, dims/strides in elements. stride1 unused (=0).
__device__ __forceinline__ v8i tdm_group1_2d(unsigned tdim0, unsigned tdim1,
                                             unsigned tile0, unsigned tile1,
                                             unsigned long long stride0) {
    v8i g;
    g[0] = (int)(2u << 16);                                   // data_size=4B
    g[1] = (int)((tdim0 & 0xFFFFu) << 16);                    // tensor_dim0 lo
    g[2] = (int)((tdim0 >> 16) | ((tdim1 & 0xFFFFu) << 16));  // dim0 hi | dim1 lo
    g[3] = (int)((tdim1 >> 16) | (tile0 << 16));              // dim1 hi | tile_dim0
    g[4] = (int)(tile1 & 0xFFFFu);                            // tile_dim1, tile_dim2=0
    g[5] = (int)(stride0 & 0xFFFFFFFFull);                    // dim0_stride lo32
    g[6] = (int)((stride0 >> 32) & 0xFFFFull);                // dim0_stride hi16
    g[7] = 0;
    return g;
}

__device__ __forceinline__ void tdm_load_to_lds(v4u g0, v8i g1) {
    v4i z4 = {0, 0, 0, 0};
#if __has_include(<hip/amd_detail/amd_gfx1250_TDM.h>)
    v8i z8 = {0, 0, 0, 0, 0, 0, 0, 0};
    __builtin_amdgcn_tensor_load_to_lds(g0, g1, z4, z4, z8, 0);
#else
    __builtin_amdgcn_tensor_load_to_lds(g0, g1, z4, z4, 0);
#endif
}

// ---------------------------------------------------------------------------
// Organ head: out[row] = sigmoid( relu(z_row @ Wo1 + bo1) @ Wo2 + bo2 )
// ---------------------------------------------------------------------------
__global__ void organ_head_kernel(const float* __restrict__ z,
                                  const float* __restrict__ Wo1,
                                  const float* __restrict__ bo1,
                                  const float* __restrict__ Wo2,
                                  const float* __restrict__ bo2,
                                  float* __restrict__ out) {
    __shared__ float zrow[D_MODEL];
    __shared__ float red[HIDDEN];
    const int row = blockIdx.x;          // 0 .. B*N-1
    const int t = threadIdx.x;           // 0 .. 255
    if (t < D_MODEL) zrow[t] = z[row * D_MODEL + t];
    __syncthreads();
    float acc = bo1[t];
#pragma unroll 8
    for (int d = 0; d < D_MODEL; ++d)
        acc = fmaf(zrow[d], Wo1[d * HIDDEN + t], acc);
    red[t] = fmaxf(acc, 0.0f) * Wo2[t];
    __syncthreads();
    for (int s = HIDDEN / 2; s > 0; s >>= 1) {
        if (t < s) red[t] += red[t + s];
        __syncthreads();
    }
    if (t == 0) out[row] = sigmoidf_(red[0] + bo2[0]);
}

// ---------------------------------------------------------------------------
// pi/pj projection GEMM: P(1024x256) = z(1024x128) @ W(128x256)
// One wave per 16x16 output tile, K stepped by 4 with V_WMMA_F32_16X16X4_F32.
// ---------------------------------------------------------------------------
__global__ __launch_bounds__(32)
void proj_gemm_kernel(const float* __restrict__ z,
                      const float* __restrict__ We_i,
                      const float* __restrict__ We_j,
                      float* __restrict__ pi,
                      float* __restrict__ pj) {
    const int mt = blockIdx.x;                 // 0..63  row tiles over B*N
    const int nt = blockIdx.y;                 // 0..15  col tiles over H
    const int w  = blockIdx.z;                 // 0 -> pi, 1 -> pj
    const float* __restrict__ W = w ? We_j : We_i;
    float* __restrict__ P = w ? pj : pi;

    const int lane = threadIdx.x;
    const int half = lane >> 4;
    const int l    = lane & 15;
    const int m0 = mt * 16, n0 = nt * 16;

    v8f acc = {};
    for (int k0 = 0; k0 < D_MODEL; k0 += 4) {
        const int ka = k0 + half * 2;
        v2f a, b;
        a.x = z[(m0 + l) * D_MODEL + ka];
        a.y = z[(m0 + l) * D_MODEL + ka + 1];
        b.x = W[(ka    ) * HIDDEN + n0 + l];
        b.y = W[(ka + 1) * HIDDEN + n0 + l];
        acc = __builtin_amdgcn_wmma_f32_16x16x4_f32(
            false, a, false, b, (short)0, acc, false, false);
    }
#pragma unroll
    for (int r = 0; r < 8; ++r)
        P[(m0 + r + half * 8) * HIDDEN + n0 + l] = acc[r];
}

// ---------------------------------------------------------------------------
// Edge head. Block = 256 threads (8 waves) handling (i, 128-wide j strip).
//  - TDM stages E[i, strip, 0:9] (4.6KB contiguous) and pi[b,i,:] (2 rows,
//    stride N*H) into LDS; waves wait on TENSORcnt then barrier.
//  - Each wave owns a 16-wide j tile; per 16-wide h chunk a 16x16 pe tile is
//    built with 3x V_WMMA_F32_16X16X4_F32 (K 9->12 zero pad), reused for both
//    batches, then relu-dotted against We2 per lane; one final 16-lane
//    xor-shuffle reduction per tile.
// ---------------------------------------------------------------------------
__global__ __launch_bounds__(256)
void edge_head_kernel(const float* __restrict__ E,     // (N,N,9)
                      const int*   __restrict__ mask,  // (N,N)
                      const float* __restrict__ We_e,  // (9,H)
                      const float* __restrict__ be1,   // (H)
                      const float* __restrict__ We2,   // (H,1)
                      const float* __restrict__ be2,   // (1)
                      const float* __restrict__ pi,    // (B*N,H)
                      const float* __restrict__ pj,    // (B*N,H)
                      float* __restrict__ out)         // (B,N,N)
{
    __shared__ float Ebuf[128 * EDGE_DIM];   // E[i, jstrip:jstrip+128, 0:9]
    __shared__ float Pibuf[BATCH * HIDDEN];  // pi[b, i, :] for b = 0,1

    const int i      = blockIdx.x;
    const int jstrip = blockIdx.y * 128;
    const int tid    = threadIdx.x;
    const int wave   = tid >> 5;
    const int lane   = tid & 31;
    const int half   = lane >> 4;
    const int l      = lane & 15;
    const int j0     = jstrip + wave * 16;

    if (tid < 32) {   // wave 0 issues both DMAs and waits on TENSORcnt
        v4u e0 = tdm_group0(lds_offset(Ebuf),
                            E + (size_t)(i * NNODE + jstrip) * EDGE_DIM);
        v8i e1 = tdm_group1_2d(128 * EDGE_DIM, 1, 128 * EDGE_DIM, 1,
                               128 * EDGE_DIM);
        tdm_load_to_lds(e0, e1);

        v4u p0 = tdm_group0(lds_offset(Pibuf), pi + (size_t)i * HIDDEN);
        v8i p1 = tdm_group1_2d(HIDDEN, BATCH, HIDDEN, BATCH,
                               (unsigned long long)NNODE * HIDDEN);
        tdm_load_to_lds(p0, p1);

        __builtin_amdgcn_s_wait_tensorcnt(0);
    }
    // Compiler must assume the LDS buffers were written (TDM has no pointer
    // operand visible to alias analysis).
    asm volatile("" : : "v"((const float*)Ebuf), "v"((const float*)Pibuf) : "memory");
    __syncthreads();

    // A tiles from LDS: E[i, j0+m, k], 3 K-chunks of 4 (K padded 9 -> 12).
    const float* __restrict__ Er = &Ebuf[(wave * 16 + l) * EDGE_DIM];
    v2f A0, A1, A2;
    if (half == 0) {
        A0.x = Er[0]; A0.y = Er[1];
        A1.x = Er[4]; A1.y = Er[5];
        A2.x = Er[8]; A2.y = 0.0f;
    } else {
        A0.x = Er[2]; A0.y = Er[3];
        A1.x = Er[6]; A1.y = Er[7];
        A2.x = 0.0f;  A2.y = 0.0f;
    }

    float acc0[8], acc1[8];
#pragma unroll
    for (int r = 0; r < 8; ++r) { acc0[r] = 0.0f; acc1[r] = 0.0f; }

    for (int h0 = 0; h0 < HIDDEN; h0 += 16) {
        const int hc = h0 + l;             // this lane's hidden column
        const int kb = half * 2;
        v2f B0, B1, B2;
        B0.x = We_e[(0 + kb) * HIDDEN + hc];
        B0.y = We_e[(1 + kb) * HIDDEN + hc];
        B1.x = We_e[(4 + kb) * HIDDEN + hc];
        B1.y = We_e[(5 + kb) * HIDDEN + hc];
        B2.x = (half == 0) ? We_e[8 * HIDDEN + hc] : 0.0f;   // rows 9..11 pad
        B2.y = 0.0f;

        v8f pe = {};
        pe = __builtin_amdgcn_wmma_f32_16x16x4_f32(false, A0, false, B0, (short)0, pe, false, false);
        pe = __builtin_amdgcn_wmma_f32_16x16x4_f32(false, A1, false, B1, (short)0, pe, false, false);
        pe = __builtin_amdgcn_wmma_f32_16x16x4_f32(false, A2, false, B2, (short)0, pe, false, false);

        const float b1v = be1[hc];
        const float w2v = We2[hc];
        const float pia = Pibuf[hc] + b1v;            // batch 0 (broadcast)
        const float pib = Pibuf[HIDDEN + hc] + b1v;   // batch 1
#pragma unroll
        for (int r = 0; r < 8; ++r) {
            const int jl = j0 + r + half * 8;
            const float pj0v = pj[(size_t)(0 * NNODE + jl) * HIDDEN + hc];
            const float pj1v = pj[(size_t)(1 * NNODE + jl) * HIDDEN + hc];
            const float base = pe[r];
            acc0[r] = fmaf(fmaxf(base + pia + pj0v, 0.0f), w2v, acc0[r]);
            acc1[r] = fmaf(fmaxf(base + pib + pj1v, 0.0f), w2v, acc1[r]);
        }
    }

    // Reduce across each 16-lane group (xor masks < 16 stay in-half).
#pragma unroll
    for (int r = 0; r < 8; ++r) {
#pragma unroll
        for (int off = 8; off >= 1; off >>= 1) {
            acc0[r] += __shfl_xor(acc0[r], off, 32);
            acc1[r] += __shfl_xor(acc1[r], off, 32);
        }
    }

    const float b2 = be2[0];
    if (l == 0) {   // lane 0 -> j rows 0..7, lane 16 -> rows 8..15
#pragma unroll
        for (int r = 0; r < 8; ++r) {
            const int jj = j0 + r + half * 8;
            const float m = (float)mask[i * NNODE + jj];
            out[(size_t)0 * NNODE * NNODE + i * NNODE + jj] = sigmoidf_(acc0[r] + b2) * m;
            out[(size_t)1 * NNODE * NNODE + i * NNODE + jj] = sigmoidf_(acc1[r] + b2) * m;
        }
    }
}

// ---------------------------------------------------------------------------
extern "C" void kernel_launch(void* const* d_in, const int* in_sizes, int n_in,
                              void* d_out, int out_size, void* d_ws, size_t ws_size,
                              hipStream_t stream) {
    const float* z    = (const float*)d_in[0];
    const float* E    = (const float*)d_in[1];
    const int*   mask = (const int*)  d_in[2];
    const float* Wo1  = (const float*)d_in[3];
    const float* bo1  = (const float*)d_in[4];
    const float* Wo2  = (const float*)d_in[5];
    const float* bo2  = (const float*)d_in[6];
    const float* We_i = (const float*)d_in[7];
    const float* We_j = (const float*)d_in[8];
    const float* We_e = (const float*)d_in[9];
    const float* be1  = (const float*)d_in[10];
    const float* We2  = (const float*)d_in[11];
    const float* be2  = (const float*)d_in[12];

    float* out_organ = (float*)d_out;                       // (B,N)
    float* out_edge  = (float*)d_out + BATCH * NNODE;       // (B,N,N)

    float* pi = (float*)d_ws;                               // (B*N, H)
    float* pj = pi + (size_t)BATCH * NNODE * HIDDEN;        // (B*N, H)

    organ_head_kernel<<<BATCH * NNODE, HIDDEN, 0, stream>>>(
        z, Wo1, bo1, Wo2, bo2, out_organ);

    proj_gemm_kernel<<<dim3((BATCH * NNODE) / 16, HIDDEN / 16, 2), 32, 0, stream>>>(
        z, We_i, We_j, pi, pj);

    edge_head_kernel<<<dim3(NNODE, NNODE / 128), 256, 0, stream>>>(
        E, mask, We_e, be1, We2, be2, pi, pj, out_edge);
}